// EquivariantLayerBlock_18279380812255
// MI455X (gfx1250) — compile-verified
//
#include <hip/hip_runtime.h>

#define NDIM 2048
#define NN   (2048LL*2048LL)
#define CH    16
#define NPART 15
#define NBANDS   64
#define ROWS_PER 32   // NDIM / NBANDS

typedef __attribute__((ext_vector_type(2))) float v2f;
typedef __attribute__((ext_vector_type(8))) float v8f;

// ---------- Stage 1: single pass over X -> row sums Sr[i,c], diag D[i,c], column partials ----------
__global__ void reduce_kernel(const float* __restrict__ X, float* __restrict__ SrT,
                              float* __restrict__ DT, float* __restrict__ colpart) {
    int band = blockIdx.x, c = blockIdx.y, t = threadIdx.x;
    __shared__ float red[256];
    float colacc[8];
    #pragma unroll
    for (int k = 0; k < 8; ++k) colacc[k] = 0.0f;

    for (int r = 0; r < ROWS_PER; ++r) {
        int i = band * ROWS_PER + r;
        const float* row = X + (size_t)c * NN + (size_t)i * NDIM;
        float racc = 0.0f;
        #pragma unroll
        for (int k = 0; k < 8; ++k) {
            float v = row[t + 256 * k];
            racc += v;
            colacc[k] += v;
        }
        red[t] = racc; __syncthreads();
        for (int s = 128; s > 0; s >>= 1) { if (t < s) red[t] += red[t + s]; __syncthreads(); }
        if (t == 0) SrT[i * CH + c] = red[0];
        if (t == 1) DT [i * CH + c] = row[i];    // diagonal element (L2 hit)
        __syncthreads();
    }
    #pragma unroll
    for (int k = 0; k < 8; ++k)
        colpart[((size_t)c * NBANDS + band) * NDIM + t + 256 * k] = colacc[k];
}

// ---------- Stage 1b: finalize column sums Sc[j,c] ----------
__global__ void colsum_finalize_kernel(const float* __restrict__ colpart, float* __restrict__ ScT) {
    int j = blockIdx.x * 256 + threadIdx.x, c = blockIdx.y;
    float acc = 0.0f;
    for (int band = 0; band < NBANDS; ++band)
        acc += colpart[((size_t)c * NBANDS + band) * NDIM + j];
    ScT[j * CH + c] = acc;
}

// ---------- Stage 2: trace T[c], total S[c], global constant K0[o] ----------
__global__ void scalars_kernel(const float* __restrict__ DT, const float* __restrict__ SrT,
                               const float* __restrict__ W, const float* __restrict__ bias,
                               float* __restrict__ Tg, float* __restrict__ Sg,
                               float* __restrict__ K0) {
    __shared__ float sT[CH], sS[CH];
    int t = threadIdx.x;
    if (t < CH) {
        float tt = 0.0f, ss = 0.0f;
        for (int i = 0; i < NDIM; ++i) { tt += DT[i * CH + t]; ss += SrT[i * CH + t]; }
        sT[t] = tt; sS[t] = ss; Tg[t] = tt; Sg[t] = ss;
    }
    __syncthreads();
    if (t < CH) {
        const float invN = 1.0f / NDIM, invN2 = invN * invN;
        float bs = 0.0f;
        for (int k = 0; k < NPART; ++k) bs += bias[k];
        float k0 = bs;  // Y += sum(bias)
        for (int c = 0; c < CH; ++c) {
            k0 += W[11*256 + c*16 + t] * sT[c] * invN     // p11: Tr/N broadcast
                + W[14*256 + c*16 + t] * sS[c] * invN2;   // p14: S/N^2 broadcast
        }
        K0[t] = k0;
    }
}

// ---------- Stage 3: per-index tables R[o,i], Cl[o,i], E[o,i] ----------
__global__ void tables_kernel(const float* __restrict__ DT, const float* __restrict__ SrT,
                              const float* __restrict__ ScT, const float* __restrict__ W,
                              const float* __restrict__ Tg, const float* __restrict__ Sg,
                              float* __restrict__ R, float* __restrict__ Cl, float* __restrict__ E) {
    int i = blockIdx.x * 256 + threadIdx.x;
    int o = blockIdx.y;
    const float invN = 1.0f / NDIM, invN2 = invN * invN;
    float r = 0.0f, cl = 0.0f, e = 0.0f, ed = 0.0f;
    for (int c = 0; c < CH; ++c) {
        float d  = DT [i * CH + c];
        float sr = SrT[i * CH + c] * invN;
        float sc = ScT[i * CH + c] * invN;
        r  += W[ 5*256 + c*16 + o] * d + W[12*256 + c*16 + o] * sr + W[ 7*256 + c*16 + o] * sc;
        cl += W[ 9*256 + c*16 + o] * d + W[13*256 + c*16 + o] * sr + W[10*256 + c*16 + o] * sc;
        e  += W[ 0*256 + c*16 + o] * d + W[ 3*256 + c*16 + o] * sr + W[ 1*256 + c*16 + o] * sc;
        ed += W[ 2*256 + c*16 + o] * Tg[c] * invN + W[ 4*256 + c*16 + o] * Sg[c] * invN2;
    }
    R [o * NDIM + i] = r;
    Cl[o * NDIM + i] = cl;
    E [o * NDIM + i] = e + ed;   // added only on the diagonal a==b
}

// ---------- Stage 4: WMMA tile computation ----------
// tI holds the identity tile [c][row][pixel] (pitch 17), tT the mirrored tile.
// Wave w computes Y[o, a0+w, b0..b0+15] via 8 chained V_WMMA_F32_16X16X4_F32.
__device__ __forceinline__ void compute_tile(
        const float* __restrict__ tI, const float* __restrict__ tT,
        const float* __restrict__ Wid, const float* __restrict__ Wtr,
        const float* __restrict__ R, const float* __restrict__ Cl,
        const float* __restrict__ E, const float* __restrict__ K0,
        float* __restrict__ Y, int a0, int b0, int w, int half, int col) {
    const int aRow = a0 + w, bCol = b0 + col;
    v8f acc;
    #pragma unroll
    for (int r = 0; r < 8; ++r) {                 // D layout: VGPR r -> M = r + 8*half
        int o = r + 8 * half;
        float v = R[o * NDIM + aRow] + Cl[o * NDIM + bCol] + K0[o];
        if (aRow == bCol) v += E[o * NDIM + aRow];
        acc[r] = v;
    }
    #pragma unroll
    for (int kk = 0; kk < 4; ++kk) {              // identity: Wid^T * X[:, a, b-tile]
        int c0 = 4 * kk + 2 * half;               // K index per lane-half (ISA 16x4 layout)
        v2f a, b;
        a.x = Wid[ c0      * 16 + col];
        a.y = Wid[(c0 + 1) * 16 + col];
        b.x = tI[ c0      * 272 + w * 17 + col];
        b.y = tI[(c0 + 1) * 272 + w * 17 + col];
        acc = __builtin_amdgcn_wmma_f32_16x16x4_f32(false, a, false, b, (short)0, acc, false, false);
    }
    #pragma unroll
    for (int kk = 0; kk < 4; ++kk) {              // transpose: Wtr^T * X[:, b-tile, a]
        int c0 = 4 * kk + 2 * half;
        v2f a, b;
        a.x = Wtr[ c0      * 16 + col];
        a.y = Wtr[(c0 + 1) * 16 + col];
        b.x = tT[ c0      * 272 + col * 17 + w];
        b.y = tT[(c0 + 1) * 272 + col * 17 + w];
        acc = __builtin_amdgcn_wmma_f32_16x16x4_f32(false, a, false, b, (short)0, acc, false, false);
    }
    #pragma unroll
    for (int r = 0; r < 8; ++r) {
        int o = r + 8 * half;
        Y[(size_t)o * NN + (size_t)aRow * NDIM + bCol] = acc[r];
    }
}

// One block per unordered tile pair (bx >= by): stage X(A,B) and X(B,A) once,
// produce both Y(A,B) and Y(B,A) -> X is read once per output tile.
__global__ __launch_bounds__(512) void equi_main_kernel(
        const float* __restrict__ X, const float* __restrict__ W,
        const float* __restrict__ R, const float* __restrict__ Cl,
        const float* __restrict__ E, const float* __restrict__ K0,
        float* __restrict__ Y) {
    const int bx = blockIdx.x, by = blockIdx.y;
    if (bx < by) return;                          // uniform exit: EXEC stays all-ones elsewhere
    __shared__ float tAB[CH * 272];               // [c][i][j] = X[c, a0+i, b0+j], pitch 17
    __shared__ float tBA[CH * 272];               // [c][j][i] = X[c, b0+j, a0+i]
    const int t  = threadIdx.x;
    const int a0 = by * 16, b0 = bx * 16;

    for (int e = t; e < CH * 64; e += 512) {      // float4-vectorized staging (b128 loads)
        int c = e >> 6, r = (e >> 2) & 15, q = (e & 3) * 4;
        float4 vAB = *(const float4*)(X + (size_t)c * NN + (size_t)(a0 + r) * NDIM + b0 + q);
        float4 vBA = *(const float4*)(X + (size_t)c * NN + (size_t)(b0 + r) * NDIM + a0 + q);
        float* dA = &tAB[c * 272 + r * 17 + q];
        dA[0] = vAB.x; dA[1] = vAB.y; dA[2] = vAB.z; dA[3] = vAB.w;
        float* dB = &tBA[c * 272 + r * 17 + q];
        dB[0] = vBA.x; dB[1] = vBA.y; dB[2] = vBA.z; dB[3] = vBA.w;
    }
    __syncthreads();

    const int w = t >> 5, lane = t & 31, half = lane >> 4, col = lane & 15;
    const float* Wid = W + 8 * 256;               // partition 8 = identity
    const float* Wtr = W + 6 * 256;               // partition 6 = transpose

    compute_tile(tAB, tBA, Wid, Wtr, R, Cl, E, K0, Y, a0, b0, w, half, col);
    if (bx != by)                                 // mirrored output tile (uniform branch)
        compute_tile(tBA, tAB, Wid, Wtr, R, Cl, E, K0, Y, b0, a0, w, half, col);
}

extern "C" void kernel_launch(void* const* d_in, const int* in_sizes, int n_in,
                              void* d_out, int out_size, void* d_ws, size_t ws_size,
                              hipStream_t stream) {
    const float* X    = (const float*)d_in[0];   // (1,16,N,N) f32
    const float* W    = (const float*)d_in[1];   // (15,16,16) f32
    const float* bias = (const float*)d_in[2];   // (15,)      f32
    float* Y = (float*)d_out;                    // (1,16,N,N) f32

    // Workspace layout (floats)
    float* ws = (float*)d_ws;
    const size_t NC = (size_t)NDIM * CH;
    float* SrT     = ws;                 // [N][16]
    float* ScT     = SrT + NC;           // [N][16]
    float* DT      = ScT + NC;           // [N][16]
    float* R       = DT  + NC;           // [16][N]
    float* Cl      = R   + NC;           // [16][N]
    float* E       = Cl  + NC;           // [16][N]
    float* Tg      = E   + NC;           // [16]
    float* Sg      = Tg  + CH;           // [16]
    float* K0      = Sg  + CH;           // [16]
    float* colpart = K0  + CH;           // [16][NBANDS][N]  (~8 MiB)

    reduce_kernel         <<<dim3(NBANDS, CH),     256, 0, stream>>>(X, SrT, DT, colpart);
    colsum_finalize_kernel<<<dim3(NDIM / 256, CH), 256, 0, stream>>>(colpart, ScT);
    scalars_kernel        <<<dim3(1),               64, 0, stream>>>(DT, SrT, W, bias, Tg, Sg, K0);
    tables_kernel         <<<dim3(NDIM / 256, CH), 256, 0, stream>>>(DT, SrT, ScT, W, Tg, Sg, R, Cl, E);
    equi_main_kernel      <<<dim3(NDIM / 16, NDIM / 16), 512, 0, stream>>>(X, W, R, Cl, E, K0, Y);
}